// SoftclDiceLoss_61838939128150
// MI455X (gfx1250) — compile-verified
//
#include <hip/hip_runtime.h>
#include <cstdint>

// Problem constants (match reference: B=8, C=1, H=W=1024, ITERS=3, SMOOTH=1)
#define HH   1024
#define WW   1024
#define NB   8
#define TILE 64
#define RAD  5                         // dependence radius of 3-iter soft_skel
#define NN   (TILE + 2 * RAD)          // 74
#define NN2  (NN * NN)                 // 5476
#define NT   256                       // threads per block (8 wave32)
#define NITER ((NN2 + NT - 1) / NT)    // 22 strided steps to cover the region
#define NBLK  ((WW / TILE) * (HH / TILE) * NB)  // 16*16*8 = 2048 tiles
#define PPT  (TILE * TILE / NT)        // 16 output pixels per thread

// ---------------------------------------------------------------------------
// CDNA5 async global->LDS copy (ASYNCcnt-tracked), per CDNA5 ISA
// ---------------------------------------------------------------------------
__device__ __forceinline__ void async_g2l_b32(uint32_t lds_off, uint64_t gaddr) {
  asm volatile("global_load_async_to_lds_b32 %0, %1, off"
               :
               : "v"(lds_off), "v"(gaddr)
               : "memory");
}
__device__ __forceinline__ void wait_async0() {
  asm volatile("s_wait_asynccnt 0" ::: "memory");
}

// ---------------------------------------------------------------------------
// Load a 74x74 halo'd tile from global into LDS via async-to-LDS.
// INT (interior) path: whole region in-image -> unpredicated, full-EXEC issue.
// Border path: out-of-image slots left untouched (garbage) -- never read
// downstream because stencils guard neighbor reads with global-coord checks
// (exactly the +/-inf padding semantics of the reference).
// ---------------------------------------------------------------------------
template <bool INT>
__device__ __forceinline__ void load_tile_async_t(const float* __restrict__ src,
                                                  float* dstLDS,
                                                  int i0, int j0, int gx0, int gy0) {
  int k = (int)threadIdx.x, i = i0, j = j0;
#pragma unroll 1
  for (int it = 0; it < NITER; ++it) {
    if (k < NN2) {
      const int gy = gy0 + i, gx = gx0 + j;
      if (INT || ((unsigned)gy < (unsigned)HH && (unsigned)gx < (unsigned)WW)) {
        const uint32_t loff = (uint32_t)(uintptr_t)(dstLDS + k);  // low 32b of AS3 addr = LDS offset
        const uint64_t ga   = (uint64_t)(uintptr_t)(src + (size_t)gy * WW + gx);
        async_g2l_b32(loff, ga);
      }
    }
    // advance flat index and (i,j) incrementally: 256 = 3*74 + 34
    k += NT; i += 3; j += 34;
    if (j >= NN) { j -= NN; ++i; }
  }
  wait_async0();
  __syncthreads();
}

// ---------------------------------------------------------------------------
// Plus-shaped 5-point erosion over valid box [lo,hi) x [lo,hi).
// INT path: no image-boundary guards (tile+ring fully interior).
// ---------------------------------------------------------------------------
template <bool INT>
__device__ __forceinline__ void erode_stage_t(const float* s, float* d,
                                              int lo, int hi,
                                              int i0, int j0, int gx0, int gy0) {
  int k = (int)threadIdx.x, i = i0, j = j0;
#pragma unroll 1
  for (int it = 0; it < NITER; ++it) {
    if (k < NN2 && (unsigned)(i - lo) < (unsigned)(hi - lo)
                && (unsigned)(j - lo) < (unsigned)(hi - lo)) {
      float v  = s[k];
      const float up = s[k - NN], dn = s[k + NN], lf = s[k - 1], rt = s[k + 1];
      if (INT) {
        v = fminf(v, up); v = fminf(v, dn);
        v = fminf(v, lf); v = fminf(v, rt);
      } else {
        const int gy = gy0 + i, gx = gx0 + j;
        v = (gy > 0)      ? fminf(v, up) : v;
        v = (gy < HH - 1) ? fminf(v, dn) : v;
        v = (gx > 0)      ? fminf(v, lf) : v;
        v = (gx < WW - 1) ? fminf(v, rt) : v;
      }
      d[k] = v;
    }
    k += NT; i += 3; j += 34;
    if (j >= NN) { j -= NN; ++i; }
  }
  __syncthreads();
}

// ---------------------------------------------------------------------------
// delta = relu(img - dilation(e)) at tile centers; skeleton update.
// INIT:  skel  = delta      !INIT: skel += relu(delta - skel*delta)
// PPT loop stays unrolled: skel[] lives in registers (static indexing).
// ---------------------------------------------------------------------------
template <bool INT, bool INIT>
__device__ __forceinline__ void delta_stage_t(const float* img, const float* e,
                                              float* skel, int gxb, int gyb) {
  const int c  = (int)threadIdx.x & (TILE - 1);
  const int rb = ((int)threadIdx.x >> 6) * PPT;
#pragma unroll
  for (int p = 0; p < PPT; ++p) {
    const int r = rb + p;
    const int k = (RAD + r) * NN + (RAD + c);
    float dv = e[k];
    if (INT) {
      dv = fmaxf(dv, e[k - NN]); dv = fmaxf(dv, e[k + NN]);
      dv = fmaxf(dv, e[k - 1]);  dv = fmaxf(dv, e[k + 1]);
    } else {
      const int gy = gyb + r, gx = gxb + c;
      dv = (gy > 0)      ? fmaxf(dv, e[k - NN]) : dv;
      dv = (gy < HH - 1) ? fmaxf(dv, e[k + NN]) : dv;
      dv = (gx > 0)      ? fmaxf(dv, e[k - 1])  : dv;
      dv = (gx < WW - 1) ? fmaxf(dv, e[k + 1])  : dv;
    }
    const float delta = fmaxf(img[k] - dv, 0.0f);
    if (INIT) skel[p] = delta;
    else      skel[p] += fmaxf(delta - skel[p] * delta, 0.0f);
  }
  __syncthreads();
}

// 3-iteration fused soft_skel over the tile currently resident in A.
// Ping-pong: A=IN -> B=E0 -> A=E1 -> B=E2 -> A=E3.
template <bool INT>
__device__ __forceinline__ void soft_skel_tile_t(float* A, float* Bb, float* skel,
                                                 int i0, int j0, int gx0, int gy0,
                                                 int gxb, int gyb) {
  erode_stage_t<INT>(A, Bb, 1, NN - 1, i0, j0, gx0, gy0);   // E0 = ero(IN)
  delta_stage_t<INT, true >(A, Bb, skel, gxb, gyb);         // skel = relu(IN - dil(E0))
  erode_stage_t<INT>(Bb, A, 2, NN - 2, i0, j0, gx0, gy0);   // E1 = ero(E0)
  delta_stage_t<INT, false>(Bb, A, skel, gxb, gyb);         // relu(E0 - dil(E1))
  erode_stage_t<INT>(A, Bb, 3, NN - 3, i0, j0, gx0, gy0);   // E2 = ero(E1)
  delta_stage_t<INT, false>(A, Bb, skel, gxb, gyb);         // relu(E1 - dil(E2))
  erode_stage_t<INT>(Bb, A, 4, NN - 4, i0, j0, gx0, gy0);   // E3 = ero(E2)
  delta_stage_t<INT, false>(Bb, A, skel, gxb, gyb);         // relu(E2 - dil(E3))
}

// Full per-tile pipeline (both images), producing the 4 partial sums.
template <bool INT>
__device__ __forceinline__ void cl_tile_body(const float* __restrict__ yp,
                                             const float* __restrict__ yt,
                                             int applySig,
                                             float* A, float* Bb,
                                             int i0, int j0, int gx0, int gy0,
                                             int gxb, int gyb, float sums[4]) {
  const int tid = (int)threadIdx.x;
  const int cc  = tid & (TILE - 1);
  const int rb0 = (tid >> 6) * PPT;

  // ---------- phase 1: skeleton of sigmoid(y_pred) ----------
  load_tile_async_t<INT>(yp, A, i0, j0, gx0, gy0);
  if (applySig) {
    int k = tid;
#pragma unroll 1
    for (int it = 0; it < NITER; ++it) {
      if (k < NN2) { const float x = A[k]; A[k] = 1.0f / (1.0f + __expf(-x)); }
      k += NT;
    }
    __syncthreads();
  }

  float sigc[PPT];  // sigmoid(y_pred) at tile centers, for the tsens numerator
#pragma unroll
  for (int p = 0; p < PPT; ++p)
    sigc[p] = A[(RAD + rb0 + p) * NN + (RAD + cc)];

  float skel[PPT];
  soft_skel_tile_t<INT>(A, Bb, skel, i0, j0, gx0, gy0, gxb, gyb);

  float s0 = 0.f, s1 = 0.f, s2 = 0.f, s3 = 0.f;
#pragma unroll
  for (int p = 0; p < PPT; ++p) s1 += skel[p];            // sum(skel_pred)

  // ---------- phase 2: skeleton of y_true ----------
  load_tile_async_t<INT>(yt, A, i0, j0, gx0, gy0);
#pragma unroll
  for (int p = 0; p < PPT; ++p)                           // sum(skel_pred * y_true)
    s0 += skel[p] * A[(RAD + rb0 + p) * NN + (RAD + cc)];

  soft_skel_tile_t<INT>(A, Bb, skel, i0, j0, gx0, gy0, gxb, gyb);
#pragma unroll
  for (int p = 0; p < PPT; ++p) {
    s3 += skel[p];            // sum(skel_true)
    s2 += skel[p] * sigc[p];  // sum(skel_true * y_pred)
  }

  sums[0] = s0; sums[1] = s1; sums[2] = s2; sums[3] = s3;
}

// ---------------------------------------------------------------------------
// Main fused kernel: one pass over both images; per-tile skeletons; 4 sums.
// grid = (16, 16, 8), block = 256.  >=4 waves/SIMD keeps VGPRs <= 256
// (no gfx1250 VGPR-MSB windowing, no spills).
// ---------------------------------------------------------------------------
__global__ __launch_bounds__(NT, 4) void cl_main(const float* __restrict__ yp,
                                                 const float* __restrict__ yt,
                                                 const int* __restrict__ flag,
                                                 float* __restrict__ partials) {
  __shared__ float A[NN2];
  __shared__ float Bb[NN2];
  __shared__ float red[NT];

  const int tid = (int)threadIdx.x;
  const int gxb = (int)blockIdx.x * TILE;
  const int gyb = (int)blockIdx.y * TILE;
  const int gx0 = gxb - RAD, gy0 = gyb - RAD;
  const size_t base = (size_t)blockIdx.z * (size_t)(HH * WW);
  const int i0 = tid / NN, j0 = tid - i0 * NN;
  const int applySig = *flag;

  // Interior <=> the 74x74 region plus its 1-ring lies inside the image,
  // so no image-boundary guard can ever trigger. True for (14/16)^2 of blocks.
  const bool interior = (gx0 >= 1) && (gy0 >= 1) &&
                        (gx0 + NN <= WW - 1) && (gy0 + NN <= HH - 1);

  float sums[4];
  if (interior)
    cl_tile_body<true >(yp + base, yt + base, applySig, A, Bb, i0, j0, gx0, gy0, gxb, gyb, sums);
  else
    cl_tile_body<false>(yp + base, yt + base, applySig, A, Bb, i0, j0, gx0, gy0, gxb, gyb, sums);

  // ---------- deterministic per-block reduction ----------
  const int blin = ((int)blockIdx.z * (HH / TILE) + (int)blockIdx.y) * (WW / TILE) + (int)blockIdx.x;
#pragma unroll 1
  for (int q = 0; q < 4; ++q) {
    __syncthreads();
    red[tid] = sums[q];
    __syncthreads();
#pragma unroll 1
    for (int off = NT / 2; off > 0; off >>= 1) {
      if (tid < off) red[tid] += red[tid + off];
      __syncthreads();
    }
    if (tid == 0) partials[q * NBLK + blin] = red[0];
  }
}

// ---------------------------------------------------------------------------
// Sigmoid-predicate prepass: flag = any(v < 0 || v > 1)   (== min<0 | max>1)
// ---------------------------------------------------------------------------
__global__ void cl_zero_flag(int* flag) { if (threadIdx.x == 0) *flag = 0; }

__global__ __launch_bounds__(NT) void cl_flag(const float4* __restrict__ yp4,
                                              int* __restrict__ flag, int n4) {
  __shared__ int f;
  if (threadIdx.x == 0) f = 0;
  __syncthreads();
  bool found = false;
#pragma unroll 1
  for (int i = (int)blockIdx.x * NT + (int)threadIdx.x; i < n4; i += (int)gridDim.x * NT) {
    const float4 v = yp4[i];
    found = found || (v.x < 0.f) || (v.x > 1.f) || (v.y < 0.f) || (v.y > 1.f)
                  || (v.z < 0.f) || (v.z > 1.f) || (v.w < 0.f) || (v.w > 1.f);
  }
  if (found) f = 1;
  __syncthreads();
  if (threadIdx.x == 0 && f) atomicOr(flag, 1);
}

// ---------------------------------------------------------------------------
// Final deterministic reduction + clDice scalar.
// ---------------------------------------------------------------------------
__global__ __launch_bounds__(NT) void cl_finalize(const float* __restrict__ partials,
                                                  float* __restrict__ out) {
  __shared__ float red[NT];
  __shared__ float sums[4];
  const int tid = (int)threadIdx.x;
#pragma unroll 1
  for (int q = 0; q < 4; ++q) {
    float a = 0.f;
#pragma unroll 1
    for (int i = tid; i < NBLK; i += NT) a += partials[q * NBLK + i];
    __syncthreads();
    red[tid] = a;
    __syncthreads();
#pragma unroll 1
    for (int off = NT / 2; off > 0; off >>= 1) {
      if (tid < off) red[tid] += red[tid + off];
      __syncthreads();
    }
    if (tid == 0) sums[q] = red[0];
    __syncthreads();
  }
  if (tid == 0) {
    const float tprec = (sums[0] + 1.0f) / (sums[1] + 1.0f);
    const float tsens = (sums[2] + 1.0f) / (sums[3] + 1.0f);
    out[0] = 1.0f - 2.0f * (tprec * tsens) / (tprec + tsens);
  }
}

// ---------------------------------------------------------------------------
extern "C" void kernel_launch(void* const* d_in, const int* in_sizes, int n_in,
                              void* d_out, int out_size, void* d_ws, size_t ws_size,
                              hipStream_t stream) {
  (void)in_sizes; (void)n_in; (void)out_size; (void)ws_size;
  const float* yp = (const float*)d_in[0];
  const float* yt = (const float*)d_in[1];
  int*   flag     = (int*)d_ws;
  float* partials = (float*)((char*)d_ws + 256);   // 4 * NBLK floats
  float* out      = (float*)d_out;

  cl_zero_flag<<<1, 64, 0, stream>>>(flag);
  cl_flag<<<1024, NT, 0, stream>>>((const float4*)yp, flag, (NB * HH * WW) / 4);

  dim3 grid(WW / TILE, HH / TILE, NB);
  cl_main<<<grid, NT, 0, stream>>>(yp, yt, flag, partials);

  cl_finalize<<<1, NT, 0, stream>>>(partials, out);
}